// VectorQuantizer_71004399337859
// MI455X (gfx1250) — compile-verified
//
#include <hip/hip_runtime.h>
#include <hip/hip_bf16.h>

typedef float v2f __attribute__((ext_vector_type(2)));
typedef float v8f __attribute__((ext_vector_type(8)));

#define NROWS   65536      // B*H*W
#define KCODES  1024
#define CDIM    64
#define ESTRIDE 68         // LDS row stride in floats (bank-conflict-free)
#define ENORM_OFF (KCODES * ESTRIDE)          // 69632 floats
#define SCR_OFF   (ENORM_OFF + KCODES)        // 70656 floats
#define SMEM_FLOATS (SCR_OFF + 8 * 16)        // + per-wave idx scratch
#define OUT_LOSS 4194304
#define OUT_PERP 4194305
#define OUT_IDX  4194306
#define OUT_MD   4259842

// ---- workspace layout (dwords) ----
// [0,1024)    histogram (u32)
// [1024,1088) z channel sums
// [1088,1152) e channel sums
// [1152,2176) e_norm
// [2176]      sum(z^2)   [2177] sum((zq-z)^2)

__global__ void vq_zero(float* ws) {
    int i = blockIdx.x * 256 + threadIdx.x;
    if (i < 2178) ws[i] = 0.0f;
}

// per-code ||e||^2 and per-channel sums of embedding
__global__ void vq_estats(const float* __restrict__ emb,
                          float* __restrict__ enorm, float* __restrict__ esum) {
    const int tid = threadIdx.x;
    for (int k = tid; k < KCODES; k += 256) {
        float s = 0.f;
        #pragma unroll 8
        for (int c = 0; c < CDIM; ++c) { float v = emb[k * CDIM + c]; s += v * v; }
        enorm[k] = s;
    }
    if (tid < CDIM) {
        float cs = 0.f;
        for (int k = 0; k < KCODES; ++k) cs += emb[k * CDIM + tid];
        esum[tid] = cs;
    }
}

// per-channel sums of z and total sum(z^2); one block per (b,c) slab of 4096
__global__ void vq_zstats(const float* __restrict__ z,
                          float* __restrict__ zsum, float* __restrict__ zsumsq) {
    __shared__ float ss[8], qq[8];
    const int tid = threadIdx.x, lane = tid & 31, wid = tid >> 5;
    const int c = blockIdx.x & 63;
    const int base = blockIdx.x * 4096;
    float s = 0.f, q = 0.f;
    for (int i = tid; i < 4096; i += 256) {
        float v = z[base + i];
        s += v; q += v * v;
    }
    #pragma unroll
    for (int o = 16; o >= 1; o >>= 1) { s += __shfl_xor(s, o, 32); q += __shfl_xor(q, o, 32); }
    if (lane == 0) { ss[wid] = s; qq[wid] = q; }
    __syncthreads();
    if (tid == 0) {
        float S = 0.f, Q = 0.f;
        #pragma unroll
        for (int i = 0; i < 8; ++i) { S += ss[i]; Q += qq[i]; }
        atomicAdd(&zsum[c], S);
        atomicAdd(zsumsq, Q);
    }
}

// main WMMA kernel: 8 waves/block, one 16-row z tile per wave, full codebook in LDS
__global__ __launch_bounds__(256) void vq_main(
        const float* __restrict__ z, const float* __restrict__ emb,
        const float* __restrict__ enorm_g, float* __restrict__ zq_out,
        float* __restrict__ idx_out, unsigned int* __restrict__ hist,
        float* __restrict__ sqsum_acc) {
    extern __shared__ float smem[];
    const int tid = threadIdx.x;
    const int lane = tid & 31;
    const int wid = tid >> 5;
    const int lw = lane & 15;   // A: row-in-tile (via lanes), B/D: code column
    const int hi = lane >> 4;   // 0/1 half of wave

    // stage embedding -> LDS [1024][68] and e_norm -> LDS
    for (int i = tid; i < KCODES * (CDIM / 4); i += 256) {
        const float4 v = ((const float4*)emb)[i];
        float* dst = &smem[(i >> 4) * ESTRIDE + ((i & 15) << 2)];
        dst[0] = v.x; dst[1] = v.y; dst[2] = v.z; dst[3] = v.w;
    }
    for (int i = tid; i < KCODES; i += 256) smem[ENORM_OFF + i] = enorm_g[i];
    __syncthreads();

    const int rowBase = blockIdx.x * 128 + wid * 16;
    const int b  = rowBase >> 12;
    const int hw = rowBase & 4095;
    const int zrowoff = b * (CDIM * 4096) + hw + lw;  // + c*4096 (NCHW, 16 rows share b,h)

    // A fragments: 16x4 fp32 per k-chunk; lanes 0-15 hold K=c0,c0+1; lanes 16-31 K=c0+2,c0+3
    v2f a[16];
    #pragma unroll
    for (int kc = 0; kc < 16; ++kc) {
        const int c0 = kc * 4 + hi * 2;
        a[kc].x = z[zrowoff + c0 * 4096];
        a[kc].y = z[zrowoff + (c0 + 1) * 4096];
    }

    float minv[8]; int mini[8];
    #pragma unroll
    for (int v = 0; v < 8; ++v) { minv[v] = 3.4e38f; mini[v] = 0; }

    for (int t = 0; t < 64; ++t) {
        const int code = t * 16 + lw;
        const float* brow = &smem[code * ESTRIDE + hi * 2];
        v8f acc = {};
        #pragma unroll
        for (int kc = 0; kc < 16; ++kc) {
            const float2 bv = *(const float2*)(brow + kc * 4);
            v2f bf; bf.x = bv.x; bf.y = bv.y;
            acc = __builtin_amdgcn_wmma_f32_16x16x4_f32(
                false, a[kc], false, bf, (short)0, acc, false, false);
        }
        const float en = smem[ENORM_OFF + code];
        #pragma unroll
        for (int v = 0; v < 8; ++v) {
            const float s = en - 2.0f * acc[v];   // ||e||^2 - 2 z.e  (||z||^2 row-const)
            if (s < minv[v]) { minv[v] = s; mini[v] = code; }
        }
    }

    // reduce (min,idx) across the 16 code-lanes of each half; ties -> lower idx
    #pragma unroll
    for (int off = 1; off < 16; off <<= 1) {
        #pragma unroll
        for (int v = 0; v < 8; ++v) {
            const float ov = __shfl_xor(minv[v], off, 32);
            const int   oi = __shfl_xor(mini[v], off, 32);
            if (ov < minv[v] || (ov == minv[v] && oi < mini[v])) { minv[v] = ov; mini[v] = oi; }
        }
    }

    // per-row winning index -> LDS scratch (rows 0..7 low half, 8..15 high half)
    float* scr = &smem[SCR_OFF + wid * 16];
    if (lw == 0) {
        #pragma unroll
        for (int v = 0; v < 8; ++v) scr[hi * 8 + v] = (float)mini[v];
    }
    __syncthreads();

    const int myIdx = (int)scr[lw];
    if (lane < 16) {
        idx_out[rowBase + lane] = (float)myIdx;
        atomicAdd(&hist[myIdx], 1u);
    }

    // z_q gather from LDS codebook + (zq - z)^2 accumulation; writes NCHW coalesced
    float lsum = 0.f;
    const float* crow = &smem[myIdx * ESTRIDE];
    #pragma unroll 4
    for (int c2 = 0; c2 < 32; ++c2) {
        const int c = c2 * 2 + hi;
        const float q = crow[c];
        const int off = zrowoff + c * 4096;
        const float zv = z[off];
        zq_out[off] = q;
        const float d = q - zv;
        lsum += d * d;
    }
    #pragma unroll
    for (int o = 16; o >= 1; o >>= 1) lsum += __shfl_xor(lsum, o, 32);
    if (lane == 0) atomicAdd(sqsum_acc, lsum);
}

__global__ __launch_bounds__(1024) void vq_final(
        const unsigned int* __restrict__ hist, const float* __restrict__ zsum,
        const float* __restrict__ esum, const float* __restrict__ enorm,
        const float* __restrict__ zsumsq, const float* __restrict__ sqsum,
        float* __restrict__ out) {
    __shared__ float r1[1024], r2[1024];
    const int tid = threadIdx.x;
    const float em = (float)hist[tid] * (1.0f / 65536.0f);
    r1[tid] = em * logf(em + 1e-10f);
    r2[tid] = enorm[tid];
    __syncthreads();
    for (int s = 512; s > 0; s >>= 1) {
        if (tid < s) { r1[tid] += r1[tid + s]; r2[tid] += r2[tid + s]; }
        __syncthreads();
    }
    if (tid == 0) {
        out[OUT_PERP] = expf(-r1[0]);
        out[OUT_LOSS] = 1.25f * sqsum[0] * (1.0f / (65536.0f * 64.0f));
        float cross = 0.f;
        #pragma unroll
        for (int c = 0; c < CDIM; ++c) cross += zsum[c] * esum[c];
        out[OUT_MD] = (1024.0f * zsumsq[0] + 65536.0f * r2[0] - 2.0f * cross)
                      * (1.0f / (65536.0f * 1024.0f));
    }
}

extern "C" void kernel_launch(void* const* d_in, const int* in_sizes, int n_in,
                              void* d_out, int out_size, void* d_ws, size_t ws_size,
                              hipStream_t stream) {
    const float* z   = (const float*)d_in[0];
    const float* emb = (const float*)d_in[1];
    float* out = (float*)d_out;
    float* ws  = (float*)d_ws;

    unsigned int* hist = (unsigned int*)ws;
    float* zsum   = ws + 1024;
    float* esum   = ws + 1088;
    float* enorm  = ws + 1152;
    float* zsumsq = ws + 2176;
    float* sqsum  = ws + 2177;

    vq_zero<<<9, 256, 0, stream>>>(ws);
    vq_estats<<<1, 256, 0, stream>>>(emb, enorm, esum);
    vq_zstats<<<1024, 256, 0, stream>>>(z, zsum, zsumsq);

    const size_t smem_bytes = SMEM_FLOATS * sizeof(float);  // ~283 KB of the 320 KB WGP LDS
    vq_main<<<512, 256, smem_bytes, stream>>>(z, emb, enorm,
                                              out /*z_q*/, out + OUT_IDX,
                                              hist, sqsum);

    vq_final<<<1, 1024, 0, stream>>>(hist, zsum, esum, enorm, zsumsq, sqsum, out);
}